// GCN_841813590016
// MI455X (gfx1250) — compile-verified
//
#include <hip/hip_runtime.h>
#include <cmath>

// ---------------------------------------------------------------------------
// GCN 3-layer forward for gfx1250 (MI455X).
// GEMMs via V_WMMA_F32_16X16X4_F32 (full fp32 precision through matrix pipe).
// Aggregation via fp32 global atomics (L2-resident working set).
// ---------------------------------------------------------------------------

typedef __attribute__((ext_vector_type(2))) float v2f;
typedef __attribute__((ext_vector_type(8))) float v8f;

// ---------------- degree / normalization ----------------

__global__ void k_zero_deg(int* __restrict__ deg, int n) {
    int i = blockIdx.x * blockDim.x + threadIdx.x;
    if (i < n) deg[i] = 0;
}

__global__ void k_count_deg(const int* __restrict__ dst, int* __restrict__ deg, int e) {
    int i = blockIdx.x * blockDim.x + threadIdx.x;
    if (i < e) atomicAdd(&deg[dst[i]], 1);
}

__global__ void k_make_dinv(const int* __restrict__ deg, float* __restrict__ dinv, int n) {
    int i = blockIdx.x * blockDim.x + threadIdx.x;
    if (i < n) dinv[i] = rsqrtf((float)deg[i] + 1.0f);   // +1 = self loop; always > 0
}

// ---------------- WMMA GEMM: out[N x COUT] = A[N x 128] @ W[128 x COUT] -----

template <int COUT>
__global__ __launch_bounds__(256) void k_gemm_wmma(const float* __restrict__ A,
                                                   const float* __restrict__ W,
                                                   float* __restrict__ out,
                                                   int nrows) {
    constexpr int K   = 128;
    constexpr int KC  = 64;        // K-chunk staged in LDS
    constexpr int LDW = KC + 2;    // pad 2: keeps 8B alignment, breaks bank conflicts
    constexpr int NT  = COUT / 16; // n-tiles per wave

    __shared__ float Wlds[COUT * LDW];

    const int tid  = threadIdx.x;
    const int wave = tid >> 5;
    const int lane = tid & 31;
    const int half = lane >> 4;    // 0: lanes 0-15, 1: lanes 16-31
    const int m    = lane & 15;

    const int rowBase = blockIdx.x * 128 + wave * 16;
    const int row     = rowBase + m;
    const int rowC    = (row < nrows) ? row : (nrows - 1);   // clamp: keep EXEC all-1s

    v8f acc[NT];
#pragma unroll
    for (int t = 0; t < NT; ++t)
        acc[t] = (v8f){0.f, 0.f, 0.f, 0.f, 0.f, 0.f, 0.f, 0.f};

    for (int chunk = 0; chunk < K / KC; ++chunk) {
        // Stage W chunk transposed: Wlds[n*LDW + kk] = W[(chunk*KC+kk)*COUT + n]
        for (int idx = tid; idx < KC * COUT; idx += 256) {
            const int kk = idx / COUT;
            const int n  = idx - kk * COUT;
            Wlds[n * LDW + kk] = W[(chunk * KC + kk) * COUT + n];
        }
        __syncthreads();

#pragma unroll
        for (int kk0 = 0; kk0 < KC; kk0 += 4) {
            // A fragment (16x4 fp32): lanes 0-15 hold K=kk0,kk0+1; lanes 16-31 K=kk0+2,kk0+3
            const int kg = chunk * KC + kk0 + 2 * half;
            const v2f a = *(const v2f*)(A + (size_t)rowC * K + kg);
#pragma unroll
            for (int t = 0; t < NT; ++t) {
                const int n  = t * 16 + m;
                const v2f b = *(const v2f*)(&Wlds[n * LDW + kk0 + 2 * half]);
                acc[t] = __builtin_amdgcn_wmma_f32_16x16x4_f32(
                    /*neg_a=*/false, a, /*neg_b=*/false, b,
                    /*c_mod=*/(short)0, acc[t],
                    /*reuse_a=*/false, /*reuse_b=*/false);
            }
        }
        __syncthreads();
    }

    // D layout: VGPR v holds row M=v (lanes 0-15) / M=v+8 (lanes 16-31), col = lane&15
#pragma unroll
    for (int t = 0; t < NT; ++t) {
        const int col = t * 16 + m;
#pragma unroll
        for (int v = 0; v < 8; ++v) {
            const int r = rowBase + v + 8 * half;
            if (r < nrows) out[(size_t)r * COUT + col] = acc[t][v];
        }
    }
}

// ---------------- bias + self-loop init: out = bias + h * dinv^2 ------------

template <int C>
__global__ void k_init_self(const float* __restrict__ h, const float* __restrict__ dinv,
                            const float* __restrict__ bias, float* __restrict__ out, int n) {
    long long i = (long long)blockIdx.x * blockDim.x + threadIdx.x;
    if (i < (long long)n * C) {
        const int r = (int)(i / C);
        const int c = (int)(i - (long long)r * C);
        const float w = dinv[r];
        out[i] = bias[c] + h[i] * (w * w);
    }
}

// ---------------- edge scatter: out[dst] += h[src] * dinv[src]*dinv[dst] ----

template <int C>
__global__ void k_scatter(const float* __restrict__ h, const int* __restrict__ src,
                          const int* __restrict__ dst, const float* __restrict__ dinv,
                          float* __restrict__ out, int e) {
    constexpr int TPE = C / 4;  // threads per edge, 4 floats each
    const long long t  = (long long)blockIdx.x * blockDim.x + threadIdx.x;
    const int ed = (int)(t / TPE);
    const int sl = (int)(t % TPE);
    if (ed < e) {
        const int s = src[ed];
        const int d = dst[ed];
        const float w = dinv[s] * dinv[d];
        const float4 v = *(const float4*)(h + (size_t)s * C + sl * 4);
        float* o = out + (size_t)d * C + sl * 4;
        unsafeAtomicAdd(o + 0, v.x * w);
        unsafeAtomicAdd(o + 1, v.y * w);
        unsafeAtomicAdd(o + 2, v.z * w);
        unsafeAtomicAdd(o + 3, v.w * w);
    }
}

// ---------------- exact GELU ------------------------------------------------

__global__ void k_gelu(float* __restrict__ x, long long n) {
    long long i = (long long)blockIdx.x * blockDim.x + threadIdx.x;
    if (i < n) {
        const float v = x[i];
        x[i] = 0.5f * v * (1.0f + erff(v * 0.70710678118654752f));
    }
}

// ---------------- launch ----------------------------------------------------

extern "C" void kernel_launch(void* const* d_in, const int* in_sizes, int n_in,
                              void* d_out, int out_size, void* d_ws, size_t ws_size,
                              hipStream_t stream) {
    const float* x  = (const float*)d_in[0];
    const int*   ei = (const int*)d_in[1];
    const float* W1 = (const float*)d_in[2];
    const float* b1 = (const float*)d_in[3];
    const float* W2 = (const float*)d_in[4];
    const float* b2 = (const float*)d_in[5];
    const float* W3 = (const float*)d_in[6];
    const float* b3 = (const float*)d_in[7];
    float* out = (float*)d_out;

    const int N = in_sizes[0] / 128;   // 50000
    const int E = in_sizes[1] / 2;     // 800000
    const int* srcI = ei;              // edge_index[0]
    const int* dstI = ei + E;          // edge_index[1]

    // workspace layout: deg(int N) | dinv(float N) | bufA(N*128) | bufB(N*128)
    char*  ws   = (char*)d_ws;
    int*   deg  = (int*)ws;
    float* dinv = (float*)(ws + (size_t)N * 4);
    float* bufA = (float*)(ws + (size_t)N * 8);
    float* bufB = bufA + (size_t)N * 128;

    const dim3 blk(256);
    const int gN   = (N + 255) / 256;
    const int gE   = (E + 255) / 256;
    const int gRow = (N + 127) / 128;                       // gemm blocks
    const int gNH  = (int)(((long long)N * 128 + 255) / 256);
    const int gNO  = (int)(((long long)N * 64 + 255) / 256);
    const int gS128 = (int)(((long long)E * 32 + 255) / 256);
    const int gS64  = (int)(((long long)E * 16 + 255) / 256);

    // normalization
    k_zero_deg<<<gN, blk, 0, stream>>>(deg, N);
    k_count_deg<<<gE, blk, 0, stream>>>(dstI, deg, E);
    k_make_dinv<<<gN, blk, 0, stream>>>(deg, dinv, N);

    // layer 1
    k_gemm_wmma<128><<<gRow, blk, 0, stream>>>(x, W1, bufA, N);
    k_init_self<128><<<gNH, blk, 0, stream>>>(bufA, dinv, b1, bufB, N);
    k_scatter<128><<<gS128, blk, 0, stream>>>(bufA, srcI, dstI, dinv, bufB, E);
    k_gelu<<<gNH, blk, 0, stream>>>(bufB, (long long)N * 128);

    // layer 2
    k_gemm_wmma<128><<<gRow, blk, 0, stream>>>(bufB, W2, bufA, N);
    k_init_self<128><<<gNH, blk, 0, stream>>>(bufA, dinv, b2, bufB, N);
    k_scatter<128><<<gS128, blk, 0, stream>>>(bufA, srcI, dstI, dinv, bufB, E);
    k_gelu<<<gNH, blk, 0, stream>>>(bufB, (long long)N * 128);

    // layer 3 (no activation)
    k_gemm_wmma<64><<<gRow, blk, 0, stream>>>(bufB, W3, bufA, N);
    k_init_self<64><<<gNO, blk, 0, stream>>>(bufA, dinv, b3, out, N);
    k_scatter<64><<<gS64, blk, 0, stream>>>(bufA, srcI, dstI, dinv, out, E);
}